// DGNRNetwork_4312147165668
// MI455X (gfx1250) — compile-verified
//
#include <hip/hip_runtime.h>

typedef __attribute__((ext_vector_type(2))) float v2f;
typedef __attribute__((ext_vector_type(8))) float v8f;

#define ENC_NEG_INF 0x007FFFFFu

// ---- monotone float <-> uint encoding for atomicMax on floats ----
__device__ __forceinline__ unsigned encf(float f) {
  unsigned u = __float_as_uint(f);
  return (u & 0x80000000u) ? ~u : (u | 0x80000000u);
}
__device__ __forceinline__ float decf(unsigned u) {
  return (u & 0x80000000u) ? __uint_as_float(u & 0x7FFFFFFFu) : __uint_as_float(~u);
}

// =====================================================================
// Fused encoder MLP: h = relu(relu(x@W1+b1)@W2+b2), x:[M,64] -> h:[M,32]
// One wave per 16-row tile; WMMA f32 16x16x4; layer-1 tile staged in LDS.
// =====================================================================
__global__ void encoder_kernel(const float* __restrict__ x,
                               const float* __restrict__ w1, const float* __restrict__ b1,
                               const float* __restrict__ w2, const float* __restrict__ b2,
                               float* __restrict__ hout, int M)
{
  __shared__ float lds[8 * 16 * 32];   // 8 waves x (16x32) tile = 16 KB
  const int lane = threadIdx.x & 31;
  const int wave = threadIdx.x >> 5;
  const int tilesM = (M + 15) >> 4;
  const int tm = blockIdx.x * (blockDim.x >> 5) + wave;
  const bool active = tm < tilesM;     // wave-uniform: EXEC stays all-ones
  const int half = lane >> 4;
  const int r = lane & 15;
  float* myLds = lds + wave * 16 * 32;

  if (active) {
    const int row0 = tm << 4;
    int rowA = row0 + r; if (rowA > M - 1) rowA = M - 1;   // clamp, keep EXEC full
    const float* arow = x + (size_t)rowA * 64;
    for (int t = 0; t < 2; ++t) {
      const int col0 = t << 4;
      v8f acc = {};
      #pragma unroll
      for (int kb = 0; kb < 64; kb += 4) {
        const int kk = kb + half * 2;
        v2f a; a.x = arow[kk];                   a.y = arow[kk + 1];
        v2f b; b.x = w1[kk * 32 + col0 + r];     b.y = w1[(kk + 1) * 32 + col0 + r];
        acc = __builtin_amdgcn_wmma_f32_16x16x4_f32(false, a, false, b,
                                                    (short)0, acc, false, false);
      }
      const float bb = b1[col0 + r];
      #pragma unroll
      for (int i = 0; i < 8; ++i)
        myLds[(i + 8 * half) * 32 + col0 + r] = fmaxf(acc[i] + bb, 0.0f);
    }
  }
  __syncthreads();
  if (active) {
    const int row0 = tm << 4;
    const float* arow = myLds + r * 32;
    for (int t = 0; t < 2; ++t) {
      const int col0 = t << 4;
      v8f acc = {};
      #pragma unroll
      for (int kb = 0; kb < 32; kb += 4) {
        const int kk = kb + half * 2;
        v2f a; a.x = arow[kk];                   a.y = arow[kk + 1];
        v2f b; b.x = w2[kk * 32 + col0 + r];     b.y = w2[(kk + 1) * 32 + col0 + r];
        acc = __builtin_amdgcn_wmma_f32_16x16x4_f32(false, a, false, b,
                                                    (short)0, acc, false, false);
      }
      const float bb = b2[col0 + r];
      #pragma unroll
      for (int i = 0; i < 8; ++i) {
        const int row = row0 + i + 8 * half;
        if (row < M)
          hout[(size_t)row * 32 + col0 + r] = fmaxf(acc[i] + bb, 0.0f);
      }
    }
  }
}

// =====================================================================
// Generic GEMM C = A@B + bias, one wave per 16x16 output tile.
// A:[M,Kdim] lda, B:[Kdim,Ncols] ldb (row-major), C:[M,Ncols] ldc.
// =====================================================================
__global__ void wmma_gemm_bias(const float* __restrict__ A, int lda,
                               const float* __restrict__ Bw, int ldb,
                               const float* __restrict__ bias,
                               float* __restrict__ C, int ldc,
                               int M, int Kdim, int Ncols)
{
  const int lane = threadIdx.x & 31;
  const int wave = threadIdx.x >> 5;
  const int tilesN = Ncols >> 4;
  const int tilesM = (M + 15) >> 4;
  const int tileId = blockIdx.x * (blockDim.x >> 5) + wave;
  if (tileId >= tilesM * tilesN) return;   // whole-wave return, no barrier below
  const int tm = tileId / tilesN;
  const int tn = tileId - tm * tilesN;
  const int row0 = tm << 4, col0 = tn << 4;
  const int half = lane >> 4;
  const int r = lane & 15;
  int rowA = row0 + r; if (rowA > M - 1) rowA = M - 1;
  const float* arow = A + (size_t)rowA * lda;

  v8f acc = {};
  for (int kb = 0; kb < Kdim; kb += 4) {
    const int kk = kb + half * 2;
    v2f a; a.x = arow[kk];                          a.y = arow[kk + 1];
    v2f b; b.x = Bw[(size_t)kk * ldb + col0 + r];   b.y = Bw[(size_t)(kk + 1) * ldb + col0 + r];
    acc = __builtin_amdgcn_wmma_f32_16x16x4_f32(false, a, false, b,
                                                (short)0, acc, false, false);
  }
  const float bb = bias[col0 + r];
  #pragma unroll
  for (int i = 0; i < 8; ++i) {
    const int row = row0 + i + 8 * half;
    if (row < M)
      C[(size_t)row * ldc + col0 + r] = acc[i] + bb;
  }
}

// ============================ fills ==================================
__global__ void fill_f32(float* __restrict__ p, float v, int n) {
  int i = blockIdx.x * blockDim.x + threadIdx.x;
  if (i < n) p[i] = v;
}
__global__ void fill_u32(unsigned* __restrict__ p, unsigned v, int n) {
  int i = blockIdx.x * blockDim.x + threadIdx.x;
  if (i < n) p[i] = v;
}

// ================= edge phase (TransformerConv attention) ============
// alpha[e,h] = dot(q[dst], k[src]) / sqrt(32); segment-max into amax (encoded)
__global__ void edge_alpha_kernel(const int* __restrict__ src, const int* __restrict__ dst,
                                  const float* __restrict__ q, const float* __restrict__ k,
                                  float* __restrict__ alphaBuf, unsigned* __restrict__ amax,
                                  int E_)
{
  int e = blockIdx.x * blockDim.x + threadIdx.x;
  if (e >= E_) return;
  const int s = src[e], d = dst[e];
  const float4* qd = (const float4*)(q + (size_t)d * 128);
  const float4* ks = (const float4*)(k + (size_t)s * 128);
  const float inv = 0.17677669529663687f;   // 1/sqrt(32)
  #pragma unroll
  for (int h = 0; h < 4; ++h) {
    float acc = 0.0f;
    #pragma unroll
    for (int j = 0; j < 8; ++j) {
      float4 a = qd[h * 8 + j], b = ks[h * 8 + j];
      acc += a.x * b.x + a.y * b.y + a.z * b.z + a.w * b.w;
    }
    const float al = acc * inv;
    alphaBuf[(size_t)e * 4 + h] = al;
    atomicMax(&amax[(size_t)d * 4 + h], encf(al));
  }
}

// ex = exp(alpha - amax[dst]); overwrite alphaBuf; denom[dst] += ex
__global__ void edge_softmax_kernel(const int* __restrict__ dst,
                                    float* __restrict__ alphaBuf,
                                    const unsigned* __restrict__ amax,
                                    float* __restrict__ denom, int E_)
{
  int e = blockIdx.x * blockDim.x + threadIdx.x;
  if (e >= E_) return;
  const int d = dst[e];
  #pragma unroll
  for (int h = 0; h < 4; ++h) {
    const float m = decf(amax[(size_t)d * 4 + h]);
    const float ex = expf(alphaBuf[(size_t)e * 4 + h] - m);
    alphaBuf[(size_t)e * 4 + h] = ex;
    atomicAdd(&denom[(size_t)d * 4 + h], ex);
  }
}

// accum[dst, :] += ex[e,h] * v[src, :]; one thread per (edge, 4-float chunk)
__global__ void edge_aggregate_kernel(const int* __restrict__ src, const int* __restrict__ dst,
                                      const float* __restrict__ v,
                                      const float* __restrict__ alphaBuf,
                                      float* __restrict__ accum, int E_)
{
  int gid = blockIdx.x * blockDim.x + threadIdx.x;
  if (gid >= E_ * 32) return;
  const int e = gid >> 5;
  const int t = gid & 31;
  const int h = t >> 3;
  const int c = h * 32 + (t & 7) * 4;
  const int s = src[e], d = dst[e];
  const float ex = alphaBuf[(size_t)e * 4 + h];
  const float4 vv = *(const float4*)(v + (size_t)s * 128 + c);
  float* ap = accum + (size_t)d * 128 + c;
  atomicAdd(ap + 0, ex * vv.x);
  atomicAdd(ap + 1, ex * vv.y);
  atomicAdd(ap + 2, ex * vv.z);
  atomicAdd(ap + 3, ex * vv.w);
}

// h = relu(accum / (denom + 1e-16)), in place over [N,128]
__global__ void normalize_relu_kernel(float* __restrict__ acc,
                                      const float* __restrict__ denom, int total)
{
  int i = blockIdx.x * blockDim.x + threadIdx.x;
  if (i >= total) return;
  const int node = i >> 7;
  const int h = (i & 127) >> 5;
  const float dn = denom[(size_t)node * 4 + h] + 1e-16f;
  acc[i] = fmaxf(acc[i] / dn, 0.0f);
}

__global__ void mask_inplace_kernel(float* __restrict__ h,
                                    const float* __restrict__ dm, int total)
{
  int i = blockIdx.x * blockDim.x + threadIdx.x;
  if (i >= total) return;
  h[i] *= dm[i >> 7];
}

__global__ void gather_kernel(const float* __restrict__ h, const int* __restrict__ gi,
                              float* __restrict__ xout, int Bn, int width)
{
  int t = blockIdx.x * blockDim.x + threadIdx.x;
  if (t >= Bn * width) return;
  const int b = t / width, c = t - b * width;
  xout[t] = h[(size_t)gi[b] * width + c];
}

// out[b,o] = concat(x1=h_enc[gi], x2buf, x3=h2[gi]) @ out_w + out_b
__global__ void head_kernel(const float* __restrict__ h_enc,
                            const float* __restrict__ x2buf,
                            const float* __restrict__ h2,
                            const int* __restrict__ gi,
                            const float* __restrict__ w, const float* __restrict__ b,
                            float* __restrict__ out, int Bn)
{
  int t = blockIdx.x * blockDim.x + threadIdx.x;
  if (t >= Bn * 5) return;
  const int bi = t / 5, o = t - bi * 5;
  const int n = gi[bi];
  float acc = b[o];
  const float* p0 = h_enc + (size_t)n * 32;
  #pragma unroll 8
  for (int j = 0; j < 32; ++j)  acc += p0[j] * w[j * 5 + o];
  const float* p1 = x2buf + (size_t)bi * 128;
  #pragma unroll 8
  for (int j = 0; j < 128; ++j) acc += p1[j] * w[(32 + j) * 5 + o];
  const float* p2 = h2 + (size_t)n * 128;
  #pragma unroll 8
  for (int j = 0; j < 128; ++j) acc += p2[j] * w[(160 + j) * 5 + o];
  out[t] = acc;
}

// =====================================================================
extern "C" void kernel_launch(void* const* d_in, const int* in_sizes, int n_in,
                              void* d_out, int out_size, void* d_ws, size_t ws_size,
                              hipStream_t stream)
{
  const float* x      = (const float*)d_in[0];
  const int*   eidx   = (const int*)d_in[1];
  const float* dm     = (const float*)d_in[2];
  const int*   gi     = (const int*)d_in[3];
  const float* enc_w1 = (const float*)d_in[4];  const float* enc_b1 = (const float*)d_in[5];
  const float* enc_w2 = (const float*)d_in[6];  const float* enc_b2 = (const float*)d_in[7];
  const float* c1_wq  = (const float*)d_in[8];  const float* c1_bq  = (const float*)d_in[9];
  const float* c1_wk  = (const float*)d_in[10]; const float* c1_bk  = (const float*)d_in[11];
  const float* c1_wv  = (const float*)d_in[12]; const float* c1_bv  = (const float*)d_in[13];
  const float* c2_wq  = (const float*)d_in[14]; const float* c2_bq  = (const float*)d_in[15];
  const float* c2_wk  = (const float*)d_in[16]; const float* c2_bk  = (const float*)d_in[17];
  const float* c2_wv  = (const float*)d_in[18]; const float* c2_bv  = (const float*)d_in[19];
  const float* out_w  = (const float*)d_in[20]; const float* out_b  = (const float*)d_in[21];
  float* out = (float*)d_out;

  const int N  = in_sizes[0] / 64;
  const int E  = in_sizes[1] / 2;
  const int Bn = in_sizes[3];
  const int* srcIdx = eidx;
  const int* dstIdx = eidx + E;

  // ---- workspace layout (floats) ----
  float* ws = (float*)d_ws;
  size_t off = 0;
  float*    h_enc    = ws + off; off += (size_t)N * 32;
  float*    q        = ws + off; off += (size_t)N * 128;
  float*    k        = ws + off; off += (size_t)N * 128;
  float*    v        = ws + off; off += (size_t)N * 128;
  float*    h1       = ws + off; off += (size_t)N * 128;   // conv1 accum -> output
  float*    h2       = ws + off; off += (size_t)N * 128;   // conv2 accum -> output
  float*    alphaBuf = ws + off; off += (size_t)E * 4;
  float*    denom    = ws + off; off += (size_t)N * 4;
  unsigned* amax     = (unsigned*)(ws + off); off += (size_t)N * 4;
  float*    x2buf    = ws + off; off += (size_t)Bn * 128;
  (void)ws_size; (void)n_in; (void)out_size;

  const int tilesM = (N + 15) / 16;
  const int BS = 256;
  const int nodeF = N * 128;

  // 1) encoder
  encoder_kernel<<<(tilesM + 7) / 8, BS, 0, stream>>>(x, enc_w1, enc_b1, enc_w2, enc_b2, h_enc, N);

  // 2) conv1 projections (K=32, Ncols=128)
  {
    const int tiles = tilesM * 8;
    wmma_gemm_bias<<<(tiles + 7) / 8, BS, 0, stream>>>(h_enc, 32, c1_wq, 128, c1_bq, q, 128, N, 32, 128);
    wmma_gemm_bias<<<(tiles + 7) / 8, BS, 0, stream>>>(h_enc, 32, c1_wk, 128, c1_bk, k, 128, N, 32, 128);
    wmma_gemm_bias<<<(tiles + 7) / 8, BS, 0, stream>>>(h_enc, 32, c1_wv, 128, c1_bv, v, 128, N, 32, 128);
  }
  // conv1 attention
  fill_u32<<<(N * 4 + BS - 1) / BS, BS, 0, stream>>>(amax, ENC_NEG_INF, N * 4);
  fill_f32<<<(N * 4 + BS - 1) / BS, BS, 0, stream>>>(denom, 0.0f, N * 4);
  fill_f32<<<(nodeF + BS - 1) / BS, BS, 0, stream>>>(h1, 0.0f, nodeF);
  edge_alpha_kernel    <<<(E + BS - 1) / BS, BS, 0, stream>>>(srcIdx, dstIdx, q, k, alphaBuf, amax, E);
  edge_softmax_kernel  <<<(E + BS - 1) / BS, BS, 0, stream>>>(dstIdx, alphaBuf, amax, denom, E);
  edge_aggregate_kernel<<<(E * 32 + BS - 1) / BS, BS, 0, stream>>>(srcIdx, dstIdx, v, alphaBuf, h1, E);
  normalize_relu_kernel<<<(nodeF + BS - 1) / BS, BS, 0, stream>>>(h1, denom, nodeF);

  // gather x2 before masking, then mask h1 in place (conv2 input)
  gather_kernel<<<(Bn * 128 + BS - 1) / BS, BS, 0, stream>>>(h1, gi, x2buf, Bn, 128);
  mask_inplace_kernel<<<(nodeF + BS - 1) / BS, BS, 0, stream>>>(h1, dm, nodeF);

  // 3) conv2 projections (K=128, Ncols=128)
  {
    const int tiles = tilesM * 8;
    wmma_gemm_bias<<<(tiles + 7) / 8, BS, 0, stream>>>(h1, 128, c2_wq, 128, c2_bq, q, 128, N, 128, 128);
    wmma_gemm_bias<<<(tiles + 7) / 8, BS, 0, stream>>>(h1, 128, c2_wk, 128, c2_bk, k, 128, N, 128, 128);
    wmma_gemm_bias<<<(tiles + 7) / 8, BS, 0, stream>>>(h1, 128, c2_wv, 128, c2_bv, v, 128, N, 128, 128);
  }
  // conv2 attention
  fill_u32<<<(N * 4 + BS - 1) / BS, BS, 0, stream>>>(amax, ENC_NEG_INF, N * 4);
  fill_f32<<<(N * 4 + BS - 1) / BS, BS, 0, stream>>>(denom, 0.0f, N * 4);
  fill_f32<<<(nodeF + BS - 1) / BS, BS, 0, stream>>>(h2, 0.0f, nodeF);
  edge_alpha_kernel    <<<(E + BS - 1) / BS, BS, 0, stream>>>(srcIdx, dstIdx, q, k, alphaBuf, amax, E);
  edge_softmax_kernel  <<<(E + BS - 1) / BS, BS, 0, stream>>>(dstIdx, alphaBuf, amax, denom, E);
  edge_aggregate_kernel<<<(E * 32 + BS - 1) / BS, BS, 0, stream>>>(srcIdx, dstIdx, v, alphaBuf, h2, E);
  normalize_relu_kernel<<<(nodeF + BS - 1) / BS, BS, 0, stream>>>(h2, denom, nodeF);

  // 4) output head
  head_kernel<<<(Bn * 5 + BS - 1) / BS, BS, 0, stream>>>(h_enc, x2buf, h2, gi, out_w, out_b, out, Bn);
}